// MFNLoss_77232101916941
// MI455X (gfx1250) — compile-verified
//
#include <hip/hip_runtime.h>
#include <math.h>
#include <utility>
#include <stdint.h>

#define H 1024
#define W 1024
#define BATCH 8
#define CH 3
#define PH 256
#define PW 256
#define NPLANES (BATCH * CH)   // 24

static constexpr size_t POOLED_N = (size_t)BATCH * CH * PH * PW;   // 1,572,864 floats
static constexpr int K1_BLOCKS = (W / 256) * (H / 4);              // 1024
static constexpr int K2_BLOCKS = (CH * H * W) / 256;               // 12288
static constexpr int K3_BLOCKS = (BATCH * PH * PW) / 256;          // 2048

static constexpr size_t OFF_PR = POOLED_N;
static constexpr size_t OFF_PC = OFF_PR + K1_BLOCKS;
static constexpr size_t OFF_PS = OFF_PC + K1_BLOCKS;
static constexpr size_t OFF_PP = OFF_PS + K2_BLOCKS;

// ---------------------------------------------------------------------------
// CDNA5 async global->LDS copy (tracked by ASYNCcnt).
// VDST operand is the 32-bit LDS byte address: the low 32 bits of a flat
// pointer into LDS are exactly the LDS offset (aperture occupies the high
// 32 bits per the ISA flat-address layout), so a truncating cast suffices.
// ---------------------------------------------------------------------------
__device__ __forceinline__ void async_copy16(const float* gp, const float* lp)
{
    const unsigned lds_addr = (unsigned)(uintptr_t)lp;
    asm volatile("global_load_async_to_lds_b128 %0, %1, off"
                 :: "v"(lds_addr), "v"(gp)
                 : "memory");
}

template <int N>
__device__ __forceinline__ void wait_async()
{
    asm volatile("s_wait_asynccnt %0" :: "n"(N) : "memory");
}

// compile-time for-loop (keeps register-array indices constant)
template <typename F, int... Is>
__device__ __forceinline__ void static_for_impl(F&& f, std::integer_sequence<int, Is...>)
{
    (f(std::integral_constant<int, Is>{}), ...);
}
template <int N, typename F>
__device__ __forceinline__ void static_for(F&& f)
{
    static_for_impl(f, std::make_integer_sequence<int, N>{});
}

// ---------------------------------------------------------------------------
// Kernel 1: single streaming pass over both tensors with a 4-deep
// async-to-LDS software pipeline (double... quad-buffered, 32 KB LDS).
//   - r_loss partial sums (L1)
//   - c_loss partial sums (per-pixel batch-dot / norms -> arccos)
//   - 4x4 pooled (b - a)/16 planes written to workspace (for p_loss pass)
// Thread layout: tid&3 = row within 4-row band, tid>>2 = 4-col group.
// Each thread's async copy slot == its consume slot -> no block barrier,
// only per-wave ASYNCcnt watermarks.
// ---------------------------------------------------------------------------
__global__ __launch_bounds__(256) void k1_main(const float* __restrict__ A,
                                               const float* __restrict__ B,
                                               float* __restrict__ pooled,
                                               float* __restrict__ pr,
                                               float* __restrict__ pc)
{
    __shared__ float tileA[4][1024];
    __shared__ float tileB[4][1024];

    const int tid  = threadIdx.x;
    const int r    = tid & 3;          // row in 4-row band
    const int cg   = tid >> 2;         // 0..63 column group (4 cols each)
    const int h    = blockIdx.y * 4 + r;
    const int w    = blockIdx.x * 256 + cg * 4;
    const int ph   = blockIdx.y;               // pooled row
    const int pw   = blockIdx.x * 64 + cg;     // pooled col
    const int tid4 = tid * 4;

    const size_t pix   = (size_t)h * W + w;
    const size_t plane = (size_t)H * W;

    float dotv[CH][4], na2[CH][4], nb2[CH][4];
    #pragma unroll
    for (int c = 0; c < CH; ++c)
        #pragma unroll
        for (int j = 0; j < 4; ++j) { dotv[c][j] = 0.f; na2[c][j] = 0.f; nb2[c][j] = 0.f; }
    float r_sum = 0.f;

    // prologue: stages 0..2 in flight
    static_for<3>([&](auto Pc) {
        constexpr int P = decltype(Pc)::value;
        const size_t off = (size_t)P * plane + pix;
        async_copy16(A + off, &tileA[P][tid4]);
        async_copy16(B + off, &tileB[P][tid4]);
    });

    static_for<NPLANES>([&](auto Pc) {
        constexpr int P   = decltype(Pc)::value;
        constexpr int rem = (NPLANES - 1) - P;          // stages issued beyond P
        wait_async<(rem >= 2 ? 4 : 2 * rem)>();         // stage P landed in LDS

        const float4 av = *(const float4*)&tileA[P & 3][tid4];
        const float4 bv = *(const float4*)&tileB[P & 3][tid4];

        if constexpr (P + 3 < NPLANES) {                // refill the buffer freed at P-1
            const size_t off = (size_t)(P + 3) * plane + pix;
            async_copy16(A + off, &tileA[(P + 3) & 3][tid4]);
            async_copy16(B + off, &tileB[(P + 3) & 3][tid4]);
        }

        constexpr int c = P % CH;
        const float aj[4] = {av.x, av.y, av.z, av.w};
        const float bj[4] = {bv.x, bv.y, bv.z, bv.w};
        float csum = 0.f;
        #pragma unroll
        for (int j = 0; j < 4; ++j) {
            r_sum     += fabsf(aj[j] - bj[j]);
            dotv[c][j] = fmaf(aj[j], bj[j], dotv[c][j]);
            na2[c][j]  = fmaf(aj[j], aj[j], na2[c][j]);
            nb2[c][j]  = fmaf(bj[j], bj[j], nb2[c][j]);
            csum      += bj[j] - aj[j];
        }
        // 4x4 pool: horizontal sum in csum, vertical across the 4 row-lanes
        float v = csum;
        v += __shfl_xor(v, 1, 32);
        v += __shfl_xor(v, 2, 32);
        if (r == 0)
            pooled[((size_t)P * PH + ph) * PW + pw] = v * (1.0f / 16.0f);
    });

    // per-pixel cosine -> arccos
    float c_sum = 0.f;
    #pragma unroll
    for (int j = 0; j < 4; ++j) {
        float cosv = 0.f;
        #pragma unroll
        for (int c = 0; c < CH; ++c) {
            const float na = fmaxf(sqrtf(na2[c][j]), 1e-12f);
            const float nb = fmaxf(sqrtf(nb2[c][j]), 1e-12f);
            cosv += dotv[c][j] / (na * nb);
        }
        cosv = fminf(fmaxf(cosv, -1.0f + 1e-7f), 1.0f - 1e-7f);
        c_sum += acosf(cosv);
    }

    __shared__ float sr[256];
    __shared__ float sc[256];
    sr[tid] = r_sum; sc[tid] = c_sum;
    __syncthreads();
    for (int s = 128; s > 0; s >>= 1) {
        if (tid < s) { sr[tid] += sr[tid + s]; sc[tid] += sc[tid + s]; }
        __syncthreads();
    }
    if (tid == 0) {
        const int bid = blockIdx.y * gridDim.x + blockIdx.x;
        pr[bid] = sr[0];
        pc[bid] = sc[0];
    }
}

// ---------------------------------------------------------------------------
// Kernel 2: Sobel on (a[0]-b[0]) per channel, zero pad, sum |.|
// Batch-0 planes (24 MB total) are L2-resident after kernel 1.
// ---------------------------------------------------------------------------
__global__ __launch_bounds__(256) void k2_sobel(const float* __restrict__ A,
                                                const float* __restrict__ B,
                                                float* __restrict__ ps)
{
    const int idx = blockIdx.x * 256 + threadIdx.x;   // < 3*1024*1024
    const int c   = idx >> 20;
    const int rem = idx & ((1 << 20) - 1);
    const int h   = rem >> 10;
    const int w   = rem & 1023;
    const float* Ac = A + (size_t)c * H * W;
    const float* Bc = B + (size_t)c * H * W;

    auto d = [&](int hh, int ww) -> float {
        if (hh < 0 || hh >= H || ww < 0 || ww >= W) return 0.f;
        const int o = hh * W + ww;
        return Ac[o] - Bc[o];
    };
    const float g = (d(h - 1, w - 1) - d(h - 1, w + 1))
                  + 2.0f * (d(h, w - 1) - d(h, w + 1))
                  + (d(h + 1, w - 1) - d(h + 1, w + 1));
    float s = fabsf(g);

    __shared__ float sm[256];
    sm[threadIdx.x] = s;
    __syncthreads();
    for (int st = 128; st > 0; st >>= 1) {
        if (threadIdx.x < st) sm[threadIdx.x] += sm[threadIdx.x + st];
        __syncthreads();
    }
    if (threadIdx.x == 0) ps[blockIdx.x] = sm[0];
}

// ---------------------------------------------------------------------------
// Kernel 3: directional terms on pooled diff planes.
//   d0 = g[w]-g[w-1], d1 = g[w]-g[w+1], d2 = g-r, d3 = g-b  (zero padded in w)
// ---------------------------------------------------------------------------
__global__ __launch_bounds__(256) void k3_spatial(const float* __restrict__ pooled,
                                                  float* __restrict__ pp)
{
    const int idx = blockIdx.x * 256 + threadIdx.x;   // < 8*256*256
    const int b = idx >> 16;
    const int h = (idx >> 8) & 255;
    const int w = idx & 255;

    const size_t plane = (size_t)PH * PW;
    const size_t base  = (size_t)b * CH * plane + (size_t)h * PW;

    const float prv = pooled[base + w];                 // R
    const float pg  = pooled[base + plane + w];         // G
    const float pbv = pooled[base + 2 * plane + w];     // B
    const float pgl = (w > 0)      ? pooled[base + plane + w - 1] : 0.f;
    const float pgr = (w < PW - 1) ? pooled[base + plane + w + 1] : 0.f;

    const float d0 = pg - pgl;
    const float d1 = pg - pgr;
    const float d2 = pg - prv;
    const float d3 = pg - pbv;
    float s = d0 * d0 + d1 * d1 + d2 * d2 + d3 * d3;

    __shared__ float sm[256];
    sm[threadIdx.x] = s;
    __syncthreads();
    for (int st = 128; st > 0; st >>= 1) {
        if (threadIdx.x < st) sm[threadIdx.x] += sm[threadIdx.x + st];
        __syncthreads();
    }
    if (threadIdx.x == 0) pp[blockIdx.x] = sm[0];
}

// ---------------------------------------------------------------------------
// Kernel 4: deterministic final reduction + weighting.
// ---------------------------------------------------------------------------
__global__ __launch_bounds__(256) void k4_final(const float* __restrict__ pr,
                                                const float* __restrict__ pc,
                                                const float* __restrict__ ps,
                                                const float* __restrict__ pp,
                                                float* __restrict__ out)
{
    const int tid = threadIdx.x;
    float r = 0.f, c = 0.f, s = 0.f, p = 0.f;
    for (int i = tid; i < K1_BLOCKS; i += 256) { r += pr[i]; c += pc[i]; }
    for (int i = tid; i < K2_BLOCKS; i += 256) s += ps[i];
    for (int i = tid; i < K3_BLOCKS; i += 256) p += pp[i];

    __shared__ float srm[256], scm[256], ssm[256], spm[256];
    srm[tid] = r; scm[tid] = c; ssm[tid] = s; spm[tid] = p;
    __syncthreads();
    for (int st = 128; st > 0; st >>= 1) {
        if (tid < st) {
            srm[tid] += srm[tid + st];
            scm[tid] += scm[tid + st];
            ssm[tid] += ssm[tid + st];
            spm[tid] += spm[tid + st];
        }
        __syncthreads();
    }
    if (tid == 0) {
        const float r_loss = srm[0] / (float)((size_t)BATCH * CH * H * W);
        const float c_loss = scm[0];
        const float s_loss = ssm[0];
        const float p_loss = spm[0] / (float)(BATCH * PH * PW);
        // W_C*c + W_R*r + W_P*p + W_S*s
        out[0] = 0.5f * c_loss + 1.0f * r_loss + 1.0f * p_loss + 0.1f * s_loss;
    }
}

extern "C" void kernel_launch(void* const* d_in, const int* in_sizes, int n_in,
                              void* d_out, int out_size, void* d_ws, size_t ws_size,
                              hipStream_t stream)
{
    (void)in_sizes; (void)n_in; (void)out_size; (void)ws_size;
    const float* A = (const float*)d_in[0];   // predictions
    const float* B = (const float*)d_in[1];   // targets
    float* ws  = (float*)d_ws;
    float* out = (float*)d_out;

    float* pooled = ws;
    float* pr = ws + OFF_PR;
    float* pc = ws + OFF_PC;
    float* ps = ws + OFF_PS;
    float* pp = ws + OFF_PP;

    dim3 g1(W / 256, H / 4);   // (4, 256)
    k1_main<<<g1, 256, 0, stream>>>(A, B, pooled, pr, pc);
    k2_sobel<<<K2_BLOCKS, 256, 0, stream>>>(A, B, ps);
    k3_spatial<<<K3_BLOCKS, 256, 0, stream>>>(pooled, pp);
    k4_final<<<1, 256, 0, stream>>>(pr, pc, ps, pp, out);
}